// Wide_19585050869933
// MI455X (gfx1250) — compile-verified
//
#include <hip/hip_runtime.h>

typedef __attribute__((ext_vector_type(2))) float v2f;
typedef __attribute__((ext_vector_type(8))) float v8f;

#define VOCAB 5000
#define NS 26      // sparse features
#define ND 13      // dense features

// Main kernel: one row per lane, 32 rows per wave, 256 threads/block.
// Dense matvec done with V_WMMA_F32_16X16X4_F32 (two 16-row tiles per wave,
// K padded 13->16 over four K=4 chunks). Gathers done per-lane.
// PRECONDITION: grid covers rows exactly (B multiple of 256) -> EXEC all 1s.
__global__ __launch_bounds__(256) void wide_wmma_kernel(
    const float* __restrict__ dense,    // [B,13]
    const float* __restrict__ emb,      // [26,5000]
    const float* __restrict__ cross0,   // [25e6]
    const float* __restrict__ cross1,   // [25e6]
    const float* __restrict__ w,        // [13]
    const int*   __restrict__ sidx,     // [B,26]
    float* __restrict__ out)            // [B]
{
    const int tid      = blockIdx.x * 256 + threadIdx.x;
    const int lane     = threadIdx.x & 31;
    const int waveBase = tid - lane;          // first of this wave's 32 rows
    const bool hi      = lane >= 16;          // which K-pair this half supplies
    const int  ncol    = lane & 15;           // N column inside a 16-wide tile

    // ---- dense_w into registers, zero-padded to K=16 (uniform address -> s_loads)
    float wk[16];
#pragma unroll
    for (int k = 0; k < ND; ++k) wk[k] = w[k];
    wk[13] = 0.f; wk[14] = 0.f; wk[15] = 0.f;

    // ---- WMMA dense matvec: C = A(16x4, rows all = w chunk) x B(4x16, cols = dense rows)
    // A layout (32-bit 16x4): VGPR0 = K0 (lanes 0-15) / K2 (lanes 16-31), VGPR1 = K1/K3.
    // B layout mirrors with N across lanes. Accumulate 4 chunks of K=4.
    v8f c0 = {};   // tile0: rows waveBase + 0..15
    v8f c1 = {};   // tile1: rows waveBase + 16..31
#pragma unroll
    for (int q = 0; q < 4; ++q) {
        const int kk = 4 * q + (hi ? 2 : 0);       // K index this half supplies (kk, kk+1)
        v2f a;
        a.x = hi ? wk[4 * q + 2] : wk[4 * q + 0];
        a.y = hi ? wk[4 * q + 3] : wk[4 * q + 1];

        // clamped in-bounds addresses; select 0 for padded K >= 13 (branchless)
        const int k0 = (kk     < ND) ? kk     : (ND - 1);
        const int k1 = (kk + 1 < ND) ? kk + 1 : (ND - 1);

        {   // tile 0
            const float* p = dense + (long)(waveBase + ncol) * ND;
            float v0 = p[k0], v1 = p[k1];
            v2f b;
            b.x = (kk     < ND) ? v0 : 0.f;
            b.y = (kk + 1 < ND) ? v1 : 0.f;
            c0 = __builtin_amdgcn_wmma_f32_16x16x4_f32(
                     false, a, false, b, (short)0, c0, false, false);
        }
        {   // tile 1
            const float* p = dense + (long)(waveBase + 16 + ncol) * ND;
            float v0 = p[k0], v1 = p[k1];
            v2f b;
            b.x = (kk     < ND) ? v0 : 0.f;
            b.y = (kk + 1 < ND) ? v1 : 0.f;
            c1 = __builtin_amdgcn_wmma_f32_16x16x4_f32(
                     false, a, false, b, (short)0, c1, false, false);
        }
    }
    // C/D layout: VGPR0 holds (M=0, N=lane) for lanes 0-15 and (M=8, N=lane-16) for 16-31.
    // All M rows of C are identical (A rows identical), so lane L's own row-dot is:
    const float dotv = hi ? c1[0] : c0[0];

    // ---- per-lane gathers for row tid
    const int* si = sidx + (long)tid * NS;
    int ix[NS];
#pragma unroll
    for (int i = 0; i < NS; ++i) ix[i] = si[i];

    float s = 0.f;
#pragma unroll
    for (int i = 0; i < NS; ++i) s += emb[i * VOCAB + ix[i]];   // 520KB table: L2/WGP$ resident

    // 100MB cross tables: random single-touch lines -> non-temporal so they
    // don't evict the emb table / index stream.
    const long cA = (long)ix[0] * VOCAB + ix[1];
    const long cB = (long)ix[2] * VOCAB + ix[3];
    s += __builtin_nontemporal_load(cross0 + cA);
    s += __builtin_nontemporal_load(cross1 + cB);

    out[tid] = dotv + s;
}

// Scalar tail for any rows beyond a multiple of 256 (empty for B=65536).
__global__ void wide_tail_kernel(
    const float* __restrict__ dense, const float* __restrict__ emb,
    const float* __restrict__ cross0, const float* __restrict__ cross1,
    const float* __restrict__ w, const int* __restrict__ sidx,
    float* __restrict__ out, int start, int n)
{
    int r = start + blockIdx.x * blockDim.x + threadIdx.x;
    if (r >= n) return;
    float acc = 0.f;
    const float* p = dense + (long)r * ND;
#pragma unroll
    for (int k = 0; k < ND; ++k) acc += p[k] * w[k];
    const int* si = sidx + (long)r * NS;
    int ix[NS];
#pragma unroll
    for (int i = 0; i < NS; ++i) ix[i] = si[i];
#pragma unroll
    for (int i = 0; i < NS; ++i) acc += emb[i * VOCAB + ix[i]];
    acc += __builtin_nontemporal_load(cross0 + (long)ix[0] * VOCAB + ix[1]);
    acc += __builtin_nontemporal_load(cross1 + (long)ix[2] * VOCAB + ix[3]);
    out[r] = acc;
}

extern "C" void kernel_launch(void* const* d_in, const int* in_sizes, int n_in,
                              void* d_out, int out_size, void* d_ws, size_t ws_size,
                              hipStream_t stream) {
    const float* dense  = (const float*)d_in[0];   // [B,13]
    const float* emb    = (const float*)d_in[1];   // [26,5000]
    const float* cross0 = (const float*)d_in[2];   // [25e6]
    const float* cross1 = (const float*)d_in[3];   // [25e6]
    const float* w      = (const float*)d_in[4];   // [13]
    const int*   sidx   = (const int*)d_in[5];     // [B,26]
    float* out = (float*)d_out;

    const int B = out_size;                        // output is [B,1]
    const int mainBlocks = B / 256;
    if (mainBlocks > 0) {
        wide_wmma_kernel<<<mainBlocks, 256, 0, stream>>>(
            dense, emb, cross0, cross1, w, sidx, out);
    }
    const int done = mainBlocks * 256;
    const int rem  = B - done;
    if (rem > 0) {
        wide_tail_kernel<<<(rem + 255) / 256, 256, 0, stream>>>(
            dense, emb, cross0, cross1, w, sidx, out, done, B);
    }
}